// RetrievalModel_77103252898034
// MI455X (gfx1250) — compile-verified
//
#include <hip/hip_runtime.h>
#include <hip/hip_bf16.h>

typedef __attribute__((ext_vector_type(2))) float v2f;
typedef __attribute__((ext_vector_type(8))) float v8f;

#define B_SZ 4096
#define EMB_D 64
#define HIST 50
#define NCAND 100000
#define H1_SZ 256
#define H2_SZ 128
#define CFEAT_PAD 324   // 5*64 + 3 scalars + 1 zero pad -> K % 4 == 0
#define K_TOP 100

// ---------------------------------------------------------------------------
// Feature gathering kernels
// ---------------------------------------------------------------------------
__global__ __launch_bounds__(64) void query_feat_kernel(
    const int* __restrict__ user_id, const int* __restrict__ feedback,
    const float* __restrict__ user_emb, float* __restrict__ qfeat) {
  int b = blockIdx.x;
  int d = threadIdx.x;                       // 64 threads, one per dim
  int uid = user_id[b];
  const int* hist = feedback + (size_t)uid * HIST;
  float s = 0.f;
  for (int h = 0; h < HIST; ++h)
    s += user_emb[(size_t)hist[h] * EMB_D + d];
  qfeat[(size_t)b * EMB_D + d] = s * (1.0f / (float)HIST);
}

__global__ __launch_bounds__(64) void cand_feat_kernel(
    const int* __restrict__ titles, const int* __restrict__ genres,
    const int* __restrict__ lang, const int* __restrict__ year,
    const int* __restrict__ runtime, const float* __restrict__ pop,
    const float* __restrict__ vc, const float* __restrict__ va,
    const float* __restrict__ title_emb, const float* __restrict__ genre_emb,
    const float* __restrict__ lang_emb, const float* __restrict__ year_emb,
    const float* __restrict__ runtime_emb, float* __restrict__ cf) {
  int b = blockIdx.x;
  int d = threadIdx.x;                       // 64 threads
  float t = 0.f;
  for (int i = 0; i < 16; ++i) t += title_emb[(size_t)titles[b * 16 + i] * EMB_D + d];
  float g = 0.f;
  for (int i = 0; i < 4; ++i) g += genre_emb[(size_t)genres[b * 4 + i] * EMB_D + d];
  float l = lang_emb[(size_t)lang[b] * EMB_D + d];
  float y = year_emb[(size_t)year[b] * EMB_D + d];
  float r = runtime_emb[(size_t)runtime[b] * EMB_D + d];
  float* o = cf + (size_t)b * CFEAT_PAD;
  o[0 * EMB_D + d] = t * (1.0f / 16.0f);
  o[1 * EMB_D + d] = g * (1.0f / 4.0f);
  o[2 * EMB_D + d] = l;
  o[3 * EMB_D + d] = y;
  o[4 * EMB_D + d] = r;
  if (d == 0) { o[320] = pop[b]; o[321] = vc[b]; o[322] = va[b]; o[323] = 0.f; }
}

// Pad cW1 [323,256] -> [324,256] with a zero row so GEMM K is a multiple of 4.
__global__ void pad_w_kernel(const float* __restrict__ w, float* __restrict__ wp) {
  int i = blockIdx.x * blockDim.x + threadIdx.x;
  if (i < CFEAT_PAD * H1_SZ) {
    int k = i / H1_SZ, n = i % H1_SZ;
    wp[i] = (k < 323) ? w[(size_t)k * H1_SZ + n] : 0.f;
  }
}

// ---------------------------------------------------------------------------
// Generic f32 WMMA GEMM: C[M,N] = act(A[M,K] @ W[K,N] + bias)
// One wave per 16x16 output tile, V_WMMA_F32_16X16X4_F32, K stepped by 4.
// ---------------------------------------------------------------------------
__global__ __launch_bounds__(32) void wmma_gemm_kernel(
    const float* __restrict__ A, const float* __restrict__ W,
    const float* __restrict__ bias, float* __restrict__ C,
    int M, int N, int K, int relu) {
  int m0 = blockIdx.x * 16;
  int n0 = blockIdx.y * 16;
  int lane = threadIdx.x & 31;
  int rc = lane & 15;                        // row for A, col for B/C
  int khalf = (lane >> 4) * 2;               // 0 or 2
  const float* arow = A + (size_t)(m0 + rc) * K + khalf;
  const float* wcol = W + (n0 + rc);
  v8f acc = {};
  for (int k = 0; k < K; k += 4) {
    v2f a = *(const v2f*)(arow + k);         // 8B-aligned pair
    v2f b;
    b.x = wcol[(size_t)(k + khalf) * N];
    b.y = wcol[(size_t)(k + khalf + 1) * N];
    acc = __builtin_amdgcn_wmma_f32_16x16x4_f32(false, a, false, b,
                                                (short)0, acc, false, false);
  }
  float bv = bias ? bias[n0 + rc] : 0.0f;
  int hi = (lane >> 4) * 8;
  for (int r = 0; r < 8; ++r) {
    float v = acc[r] + bv;
    if (relu) v = fmaxf(v, 0.0f);
    C[(size_t)(m0 + r + hi) * N + (n0 + rc)] = v;
  }
}

// ---------------------------------------------------------------------------
// Fused scoring + exact streaming top-100.
// Block = 256 threads (8 waves) owns 16 query rows. Per iteration:
//   1) coalesced float4 staging of 128 candidate rows into LDS (bank-padded),
//   2) each wave computes one 16x16 WMMA score tile from LDS fragments,
//   3) register-direct insertion into per-row (value,id) buffers with a
//      running 100th-value threshold; buffers compacted by in-LDS bitonic
//      sort when they pass 128 entries.  Exact top-100, never materializes
//      the 1.6 GB score matrix.
// ---------------------------------------------------------------------------
#define MT 16
#define WPB 8
#define TPB 256
#define CAP 256
#define CT_STRIDE 68          // 64 + 4 pad floats: fragment lanes hit 16 banks
#define NTILE (WPB * 16)      // 128 candidates per iteration

__device__ __forceinline__ void bitonic_sort_desc(float* v, int* ix, int n, int tid) {
  for (int i = n + tid; i < CAP; i += TPB) { v[i] = -__builtin_inff(); ix[i] = -1; }
  __syncthreads();
  for (int k = 2; k <= CAP; k <<= 1) {
    for (int j = k >> 1; j > 0; j >>= 1) {
      int i = tid;
      int ixj = i ^ j;
      if (ixj > i) {
        bool up = ((i & k) == 0);            // descending blocks
        float vi = v[i], vj = v[ixj];
        if (up ? (vi < vj) : (vi > vj)) {
          v[i] = vj; v[ixj] = vi;
          int t = ix[i]; ix[i] = ix[ixj]; ix[ixj] = t;
        }
      }
      __syncthreads();
    }
  }
}

__global__ __launch_bounds__(TPB) void score_topk_kernel(
    const float* __restrict__ q, const float* __restrict__ cands,
    int* __restrict__ preds) {
  extern __shared__ __align__(16) float smem[];
  float* ctile  = smem;                              // NTILE * CT_STRIDE floats
  float* bufv   = ctile + NTILE * CT_STRIDE;         // MT * CAP
  int*   bufi   = (int*)(bufv + MT * CAP);           // MT * CAP
  float* thresh = (float*)(bufi + MT * CAP);         // MT
  int*   cnt    = (int*)(thresh + MT);               // MT

  int tid = threadIdx.x;
  int m0 = blockIdx.x * MT;

  // Stage q tile into ctile rows 0..15 (overlaid; consumed before streaming).
  for (int i = tid; i < MT * EMB_D; i += TPB)
    ctile[(i / EMB_D) * CT_STRIDE + (i % EMB_D)] =
        q[(size_t)(m0 + i / EMB_D) * EMB_D + (i % EMB_D)];
  if (tid < MT) { thresh[tid] = -__builtin_inff(); cnt[tid] = 0; }
  __syncthreads();

  int lane = tid & 31;
  int wave = tid >> 5;
  int rc = lane & 15;
  int khalf = (lane >> 4) * 2;
  int hi = (lane >> 4) * 8;

  // Preload A fragments for all 16 k-steps (K=64) from LDS.
  v2f afrag[16];
  {
    const float* qrow = ctile + rc * CT_STRIDE + khalf;
    for (int ks = 0; ks < 16; ++ks)
      afrag[ks] = *(const v2f*)(qrow + ks * 4);
  }
  __syncthreads();                                   // ctile reused below

  const int niter = (NCAND + NTILE - 1) / NTILE;     // 782

  for (int it = 0; it < niter; ++it) {
    // ---- Stage 128 candidate rows (32KB) with coalesced b128 loads ----
    for (int f = tid; f < NTILE * (EMB_D / 4); f += TPB) {
      int row = f >> 4;                              // 16 float4 per row
      int c4 = f & 15;
      int gid = it * NTILE + row;
      float4 val = make_float4(0.f, 0.f, 0.f, 0.f);
      if (gid < NCAND)
        val = *(const float4*)(cands + (size_t)gid * EMB_D + c4 * 4);
      *(float4*)(ctile + row * CT_STRIDE + c4 * 4) = val;
    }
    __syncthreads();

    // ---- One 16x16 WMMA tile per wave, B fragments from LDS ----
    v8f acc = {};
    {
      const float* crow = ctile + (wave * 16 + rc) * CT_STRIDE + khalf;
      for (int ks = 0; ks < 16; ++ks) {
        v2f b = *(const v2f*)(crow + ks * 4);
        acc = __builtin_amdgcn_wmma_f32_16x16x4_f32(false, afrag[ks], false, b,
                                                    (short)0, acc, false, false);
      }
    }

    // ---- Register-direct insertion (lane owns candidate cid, rows r+hi) ----
    int cid = it * NTILE + wave * 16 + rc;
    if (cid < NCAND) {
      for (int r = 0; r < 8; ++r) {
        float v = acc[r];
        int row = r + hi;
        if (v > thresh[row]) {
          int pos = atomicAdd(&cnt[row], 1);
          if (pos < CAP) {
            bufv[row * CAP + pos] = v;
            bufi[row * CAP + pos] = cid;
          }
        }
      }
    }
    __syncthreads();

    // ---- Compact rows past the trigger (max +128/iter, so count <= CAP) ----
    for (int row = 0; row < MT; ++row) {
      if (cnt[row] > 128) {
        int n = cnt[row]; if (n > CAP) n = CAP;
        bitonic_sort_desc(bufv + row * CAP, bufi + row * CAP, n, tid);
        if (tid == 0) { cnt[row] = K_TOP; thresh[row] = bufv[row * CAP + K_TOP - 1]; }
        __syncthreads();
      }
    }
  }

  // Final: sort each row's surviving buffer, emit top-100 ids (desc score).
  for (int row = 0; row < MT; ++row) {
    int n = cnt[row]; if (n > CAP) n = CAP;
    bitonic_sort_desc(bufv + row * CAP, bufi + row * CAP, n, tid);
    if (tid < K_TOP)
      preds[(size_t)(m0 + row) * K_TOP + tid] = bufi[row * CAP + tid];
    __syncthreads();
  }
}

// ---------------------------------------------------------------------------
extern "C" void kernel_launch(void* const* d_in, const int* in_sizes, int n_in,
                              void* d_out, int out_size, void* d_ws, size_t ws_size,
                              hipStream_t stream) {
  const int*   user_id     = (const int*)d_in[0];
  const int*   feedback    = (const int*)d_in[1];
  const int*   titles      = (const int*)d_in[2];
  const int*   genres      = (const int*)d_in[3];
  const int*   lang        = (const int*)d_in[4];
  const int*   year        = (const int*)d_in[5];
  const int*   runtime     = (const int*)d_in[6];
  const float* pop         = (const float*)d_in[7];
  const float* vc          = (const float*)d_in[8];
  const float* va          = (const float*)d_in[9];
  const float* cands       = (const float*)d_in[10];
  const float* user_emb    = (const float*)d_in[11];
  const float* title_emb   = (const float*)d_in[12];
  const float* genre_emb   = (const float*)d_in[13];
  const float* lang_emb    = (const float*)d_in[14];
  const float* year_emb    = (const float*)d_in[15];
  const float* runtime_emb = (const float*)d_in[16];
  const float* qW1 = (const float*)d_in[17];
  const float* qb1 = (const float*)d_in[18];
  const float* qW2 = (const float*)d_in[19];
  const float* qb2 = (const float*)d_in[20];
  const float* qWo = (const float*)d_in[21];
  const float* qbo = (const float*)d_in[22];
  const float* cW1 = (const float*)d_in[23];
  const float* cb1 = (const float*)d_in[24];
  const float* cW2 = (const float*)d_in[25];
  const float* cb2 = (const float*)d_in[26];
  const float* cWo = (const float*)d_in[27];
  (void)n_in; (void)in_sizes; (void)out_size; (void)ws_size;

  // Workspace layout (floats)
  float* ws    = (float*)d_ws;
  float* qfeat = ws;                                   // 4096*64
  float* cfeat = qfeat + (size_t)B_SZ * EMB_D;         // 4096*324
  float* h1    = cfeat + (size_t)B_SZ * CFEAT_PAD;     // 4096*256
  float* h2    = h1 + (size_t)B_SZ * H1_SZ;            // 4096*128
  float* w1p   = h2 + (size_t)B_SZ * H2_SZ;            // 324*256

  float* out  = (float*)d_out;
  float* qout = out;                                   // [4096,64]
  float* cout = out + (size_t)B_SZ * EMB_D;            // [4096,64]
  int*   preds = (int*)d_out + 2 * (size_t)B_SZ * EMB_D; // [4096,100] int32

  // Feature extraction
  query_feat_kernel<<<B_SZ, 64, 0, stream>>>(user_id, feedback, user_emb, qfeat);
  cand_feat_kernel<<<B_SZ, 64, 0, stream>>>(titles, genres, lang, year, runtime,
                                            pop, vc, va, title_emb, genre_emb,
                                            lang_emb, year_emb, runtime_emb, cfeat);
  pad_w_kernel<<<(CFEAT_PAD * H1_SZ + 255) / 256, 256, 0, stream>>>(cW1, w1p);

  // Query tower MLP (all WMMA f32)
  wmma_gemm_kernel<<<dim3(B_SZ / 16, H1_SZ / 16), 32, 0, stream>>>(
      qfeat, qW1, qb1, h1, B_SZ, H1_SZ, EMB_D, 1);
  wmma_gemm_kernel<<<dim3(B_SZ / 16, H2_SZ / 16), 32, 0, stream>>>(
      h1, qW2, qb2, h2, B_SZ, H2_SZ, H1_SZ, 1);
  wmma_gemm_kernel<<<dim3(B_SZ / 16, EMB_D / 16), 32, 0, stream>>>(
      h2, qWo, qbo, qout, B_SZ, EMB_D, H2_SZ, 0);

  // Candidate tower MLP
  wmma_gemm_kernel<<<dim3(B_SZ / 16, H1_SZ / 16), 32, 0, stream>>>(
      cfeat, w1p, cb1, h1, B_SZ, H1_SZ, CFEAT_PAD, 1);
  wmma_gemm_kernel<<<dim3(B_SZ / 16, H2_SZ / 16), 32, 0, stream>>>(
      h1, cW2, cb2, h2, B_SZ, H2_SZ, H1_SZ, 1);
  wmma_gemm_kernel<<<dim3(B_SZ / 16, EMB_D / 16), 32, 0, stream>>>(
      h2, cWo, nullptr, cout, B_SZ, EMB_D, H2_SZ, 0);

  // Fused scoring (WMMA) + streaming exact top-100
  size_t smem_bytes = (size_t)(NTILE * CT_STRIDE + 2 * MT * CAP + 2 * MT) * 4 + 128;
  score_topk_kernel<<<B_SZ / MT, TPB, smem_bytes, stream>>>(qout, cands, preds);
}